// DropBlock_56255481643452
// MI455X (gfx1250) — compile-verified
//
#include <hip/hip_runtime.h>
#include <hip/hip_bf16.h>

typedef __attribute__((ext_vector_type(2))) float v2f;
typedef __attribute__((ext_vector_type(4))) float f4v;
typedef __attribute__((ext_vector_type(8))) float v8f;

// Problem constants
constexpr int Bn  = 64, Cn = 64, Hn = 112, Wn = 112;
constexpr int HW  = Hn * Wn;            // 12544
constexpr int BHW = Bn * HW;            // 802816
constexpr int CHW4 = Cn * HW / 4;       // 200704 float4 per batch image
constexpr int HW4  = HW / 4;            // 3136 float4 per mask plane
constexpr int TOT4 = Bn * CHW4;         // 12,845,056 float4 total
constexpr int NPAIR = BHW / 2;          // 401,408 threefry pairs
constexpr int NPART = 1024;             // partial-sum blocks

constexpr double GAMMA_D = (1.0 - 0.9) * (112.0 * 112.0) /
                           (7.0 * 7.0 * 106.0 * 106.0);
constexpr float KEEP_P = (float)(1.0 - GAMMA_D);   // bernoulli keep prob

// ---------------- threefry2x32-20 (key = jax.random.key(42) -> [0,42]) ----------
#define TF_ROUND(x0, x1, r)                         \
  {                                                 \
    x0 += x1;                                       \
    x1 = ((x1 << (r)) | (x1 >> (32 - (r))));        \
    x1 ^= x0;                                       \
  }

__device__ __forceinline__ void threefry20(unsigned& x0, unsigned& x1) {
  const unsigned ks0 = 0u, ks1 = 42u, ks2 = 0u ^ 42u ^ 0x1BD11BDAu;
  x0 += ks0; x1 += ks1;
  TF_ROUND(x0, x1, 13) TF_ROUND(x0, x1, 15) TF_ROUND(x0, x1, 26) TF_ROUND(x0, x1, 6)
  x0 += ks1; x1 += ks2 + 1u;
  TF_ROUND(x0, x1, 17) TF_ROUND(x0, x1, 29) TF_ROUND(x0, x1, 16) TF_ROUND(x0, x1, 24)
  x0 += ks2; x1 += ks0 + 2u;
  TF_ROUND(x0, x1, 13) TF_ROUND(x0, x1, 15) TF_ROUND(x0, x1, 26) TF_ROUND(x0, x1, 6)
  x0 += ks0; x1 += ks1 + 3u;
  TF_ROUND(x0, x1, 17) TF_ROUND(x0, x1, 29) TF_ROUND(x0, x1, 16) TF_ROUND(x0, x1, 24)
  x0 += ks1; x1 += ks2 + 4u;
  TF_ROUND(x0, x1, 13) TF_ROUND(x0, x1, 15) TF_ROUND(x0, x1, 26) TF_ROUND(x0, x1, 6)
  x0 += ks2; x1 += ks0 + 5u;
}

// ---------------- Kernel 1: fused mask generation (one block per batch image) ---
// threefry bernoulli -> row erosion (regs across barrier) -> col erosion -> global
// Single 50KB LDS plane; window [a-2, a+3], out-of-range = 1.0 (min identity).
__global__ void __launch_bounds__(1024) gen_mask(float* __restrict__ mask) {
  __shared__ float plane[HW];            // 50,176 bytes
  const int b = blockIdx.x;
  const int tid = threadIdx.x;

  // Stage 1: bernoulli init into LDS. Cell g pairs (g, g+NPAIR) across planes;
  // recompute the pair and keep the half belonging to this cell.
  for (int j = tid; j < HW; j += 1024) {
    int g = b * HW + j;
    unsigned x0, x1;
    bool lo = (g < NPAIR);
    x0 = lo ? (unsigned)g : (unsigned)(g - NPAIR);
    x1 = x0 + (unsigned)NPAIR;
    threefry20(x0, x1);
    unsigned bits = lo ? x0 : x1;
    float u = __uint_as_float((bits >> 9) | 0x3F800000u) - 1.0f;
    plane[j] = (u < KEEP_P) ? 1.0f : 0.0f;
  }
  __syncthreads();

  // Stage 2: row erosion, buffered in registers so one LDS plane suffices.
  float r[13];
  int nmine = 0;
  for (int j = tid; j < HW; j += 1024, ++nmine) {
    int w = j % Wn;
    int base = j - w;
    float m = 1.0f;
#pragma unroll
    for (int d = -2; d <= 3; ++d) {
      int ww = w + d;
      if (ww >= 0 && ww < Wn) m = fminf(m, plane[base + ww]);
    }
    r[nmine] = m;
  }
  __syncthreads();
  {
    int k = 0;
    for (int j = tid; j < HW; j += 1024, ++k) plane[j] = r[k];
  }
  __syncthreads();

  // Stage 3: column erosion -> global mask.
  for (int j = tid; j < HW; j += 1024) {
    int w = j % Wn;
    int h = j / Wn;
    float m = 1.0f;
#pragma unroll
    for (int d = -2; d <= 3; ++d) {
      int hh = h + d;
      if (hh >= 0 && hh < Hn) m = fminf(m, plane[hh * Wn + w]);
    }
    mask[b * HW + j] = m;
  }
}

// ---------------- Kernel 2: partial sums of x*mask (WMMA wave reduction) --------
__global__ void __launch_bounds__(256) partial_sum(const f4v* __restrict__ x,
                                                   const f4v* __restrict__ mask,
                                                   float* __restrict__ partials) {
  const int tid = threadIdx.x;
  const int stride = gridDim.x * blockDim.x;   // 262,144 -> exactly 49 iters/thread
  float s = 0.0f;
  for (int i = blockIdx.x * blockDim.x + tid; i < TOT4; i += stride) {
    __builtin_prefetch((const void*)(x + i + stride), 0, 0);  // global_prefetch_b8
    int b  = i / CHW4;
    int p  = i - b * CHW4;
    int hw = p % HW4;
    f4v xv = x[i];                       // RT: seed L2 for the reversed apply pass
    f4v mv = mask[b * HW4 + hw];
    f4v prod = xv * mv;
    s += prod.x + prod.y + prod.z + prod.w;
  }
  // Wave32 reduction via native-f32 WMMA: D = A(16x4 partials) x ones(4x16).
  // Lane L<16 -> A[L][0], lane L>=16 -> A[L-16][2]; row i of D = s_i + s_{i+16},
  // replicated across columns. Sum the 8 C/D VGPRs, then one xor-16 shuffle.
  float t;
#if __has_builtin(__builtin_amdgcn_wmma_f32_16x16x4_f32)
  {
    v2f a; a[0] = s;    a[1] = 0.0f;
    v2f o; o[0] = 1.0f; o[1] = 1.0f;
    v8f c = {};
    v8f d = __builtin_amdgcn_wmma_f32_16x16x4_f32(
        false, a, false, o, (short)0, c, false, false);
    t = d[0] + d[1] + d[2] + d[3] + d[4] + d[5] + d[6] + d[7];
    t += __shfl_xor(t, 16, 32);
  }
#else
  {
    t = s;
    for (int m = 16; m > 0; m >>= 1) t += __shfl_xor(t, m, 32);
  }
#endif
  __shared__ float lds[8];
  int lane = tid & 31, wave = tid >> 5;
  if (lane == 0) lds[wave] = t;
  __syncthreads();
  if (tid == 0) {
    float tot = 0.0f;
    for (int k = 0; k < 8; ++k) tot += lds[k];  // fixed order -> deterministic
    partials[blockIdx.x] = tot;
  }
}

// ---------------- Kernel 3: finalize scale = B*H*W / sum ------------------------
__global__ void __launch_bounds__(1024) finalize_scale(const float* __restrict__ partials,
                                                       float* __restrict__ scale) {
  __shared__ float sm[NPART];
  int tid = threadIdx.x;
  sm[tid] = partials[tid];
  __syncthreads();
  for (int off = NPART / 2; off > 0; off >>= 1) {
    if (tid < off) sm[tid] += sm[tid + off];
    __syncthreads();
  }
  if (tid == 0) scale[0] = (float)BHW / sm[0];
}

// ---------------- Kernel 4: out = x * mask * scale (reversed, non-temporal) -----
// Descending traversal re-reads the tail of x while it is still L2-resident from
// partial_sum; NT load/store keep the dead streams from evicting useful lines.
__global__ void __launch_bounds__(256) apply_mask(const f4v* __restrict__ x,
                                                  const f4v* __restrict__ mask,
                                                  const float* __restrict__ scalep,
                                                  f4v* __restrict__ out) {
  int gid = blockIdx.x * blockDim.x + threadIdx.x;  // exact grid, no bounds check
  int i = TOT4 - 1 - gid;                           // reverse order
  float sc = scalep[0];                             // uniform scalar load
  int b  = i / CHW4;
  int p  = i - b * CHW4;
  int hw = p % HW4;
  f4v xv = __builtin_nontemporal_load(x + i);
  f4v mv = mask[b * HW4 + hw];                      // tiny, L2-resident, keep RT
  f4v o = xv * mv * sc;
  __builtin_nontemporal_store(o, out + i);
}

extern "C" void kernel_launch(void* const* d_in, const int* in_sizes, int n_in,
                              void* d_out, int out_size, void* d_ws, size_t ws_size,
                              hipStream_t stream) {
  (void)in_sizes; (void)n_in; (void)out_size; (void)ws_size;
  const float* x = (const float*)d_in[0];
  float* out = (float*)d_out;

  float* wsf      = (float*)d_ws;
  float* scale    = wsf;                 // [0]
  float* partials = wsf + 64;            // 1024 floats
  float* maskm    = wsf + 2048;          // BHW floats (16B-aligned offset)

  gen_mask<<<Bn, 1024, 0, stream>>>(maskm);
  partial_sum<<<NPART, 256, 0, stream>>>((const f4v*)x, (const f4v*)maskm,
                                         partials);
  finalize_scale<<<1, NPART, 0, stream>>>(partials, scale);
  apply_mask<<<TOT4 / 256, 256, 0, stream>>>((const f4v*)x, (const f4v*)maskm,
                                             scale, (f4v*)out);
}